// GRUCellStacked_72318659330862
// MI455X (gfx1250) — compile-verified
//
#include <hip/hip_runtime.h>

// GRU stack: H=2, G=2 layers, T=8, B = in_sizes[0]/2 (2,097,152).
// Memory-bound (~272MB @ 23.3TB/s ~ 12us); matmuls offloaded to
// v_wmma_f32_16x16x4_f32, one WMMA per (layer, timestep) per 16 batch elems.
// Activations use v_exp_f32 + v_rcp_f32 (no IEEE divide expansion).

typedef __attribute__((ext_vector_type(2))) float v2f;
typedef __attribute__((ext_vector_type(8))) float v8f;

#define T_STEPS 8
#define N_LAYER 2

__device__ __forceinline__ float fast_rcp(float x) {
    return __builtin_amdgcn_rcpf(x);   // v_rcp_f32_e32, ~1 ULP
}
__device__ __forceinline__ float sigm(float x) {
    return fast_rcp(1.0f + __expf(-x));
}
__device__ __forceinline__ float tanh_f(float x) {
    x = fminf(fmaxf(x, -15.0f), 15.0f);
    float e = __expf(2.0f * x);
    return (e - 1.0f) * fast_rcp(e + 1.0f);
}
// Broadcast lane (L&15)'s value to lanes L and L+16 (wave32).
__device__ __forceinline__ float bcast_lo16(float v) {
    int idx = (int)((threadIdx.x & 15u) << 2);   // byte index, bits [5:2]
    int r = __builtin_amdgcn_ds_bpermute(idx, __builtin_bit_cast(int, v));
    return __builtin_bit_cast(float, r);
}

__global__ __launch_bounds__(256) void gru_fwd_kernel(
    const float* __restrict__ x_in,    // (B, 2)
    const float* __restrict__ labels,  // (B, 8, 2)
    const float* __restrict__ w_ih,    // (2, 6, 2)
    const float* __restrict__ w_hh,    // (2, 6, 2)
    const float* __restrict__ b_ih,    // (2, 6)
    float* __restrict__ ret,           // (B, 8, 2)
    float* __restrict__ block_sum)     // one partial per block
{
    const unsigned tid  = blockIdx.x * 256u + threadIdx.x;
    const unsigned lane = threadIdx.x & 31u;
    const unsigned sub  = lane & 15u;           // batch sub-index within wave
    const bool     hi   = lane >= 16u;          // upper half-wave: supplies K2..3 (h)
    const unsigned e    = (tid >> 5) * 16u + sub; // batch element (replicated on lane pair)

    // Prefetch this element's label row (read much later, after the recurrence).
    __builtin_prefetch(labels + (size_t)e * (T_STEPS * 2), 0, 0);

    // ---------------- Build per-layer A (weights) and C (bias) operands -------
    // A row m (m = sub): 0..3 fused r/z gates (w_ih@K01, w_hh@K23),
    //                    4..5 i_n (w_ih only), 6..7 h_n (w_hh rows 4..5 only),
    //                    8..15 zero.  C row m = b_ih[m] for m<6, else 0.
    v2f Aop[N_LAYER];
    v8f Cop[N_LAYER];
    const int m = (int)sub;
#pragma unroll
    for (int l = 0; l < N_LAYER; ++l) {
        const float* wi = w_ih + l * 12;
        const float* wh = w_hh + l * 12;
        const float* bi = b_ih + l * 6;
        float a0 = 0.0f, a1 = 0.0f;
        if (!hi) {                       // K = 0,1 : input weights
            if (m < 6) { a0 = wi[m * 2 + 0]; a1 = wi[m * 2 + 1]; }
        } else {                         // K = 2,3 : hidden weights
            int r = (m < 4) ? m : ((m >= 6 && m < 8) ? (m - 2) : -1);
            if (r >= 0) { a0 = wh[r * 2 + 0]; a1 = wh[r * 2 + 1]; }
        }
        v2f a; a[0] = a0; a[1] = a1;
        Aop[l] = a;
        v8f c = {};
        if (!hi) {
#pragma unroll
            for (int j = 0; j < 6; ++j) c[j] = bi[j];
        }
        Cop[l] = c;
    }

    // ---------------- Recurrent state (replicated across the lane pair) -------
    float out0 = x_in[(size_t)e * 2 + 0];
    float out1 = x_in[(size_t)e * 2 + 1];
    float h[N_LAYER][2] = {{0.0f, 0.0f}, {0.0f, 0.0f}};
    float outbuf[T_STEPS * 2];

#pragma unroll
    for (int t = 0; t < T_STEPS; ++t) {
#pragma unroll
        for (int l = 0; l < N_LAYER; ++l) {
            // B operand: lanes 0-15 carry x (layer input), lanes 16-31 carry h[l].
            v2f b;
            b[0] = hi ? h[l][0] : out0;
            b[1] = hi ? h[l][1] : out1;
            // D[m][n] for element n lands in lane n (n<16), d[0..7] = rows 0..7.
            v8f d = __builtin_amdgcn_wmma_f32_16x16x4_f32(
                false, Aop[l], false, b, (short)0, Cop[l], false, false);

            float r0 = sigm(d[0]);
            float r1 = sigm(d[1]);
            float z0 = sigm(d[2]);
            float z1 = sigm(d[3]);
            float n0 = tanh_f(d[4] + r0 * d[6]);
            float n1 = tanh_f(d[5] + r1 * d[7]);
            float hn0 = (1.0f - z0) * n0 + z0 * h[l][0];
            float hn1 = (1.0f - z1) * n1 + z1 * h[l][1];
            // Replicate new h to the upper half-wave for the next WMMA.
            hn0 = bcast_lo16(hn0);
            hn1 = bcast_lo16(hn1);
            h[l][0] = hn0; h[l][1] = hn1;
            out0 = hn0; out1 = hn1;     // layer output feeds next layer / next t
        }
        outbuf[t * 2 + 0] = out0;
        outbuf[t * 2 + 1] = out1;
    }

    // ---------------- Store ret + per-thread squared-error partial -------------
    float lsum = 0.0f;
    if (!hi) {
        const float4* lb = (const float4*)(labels + (size_t)e * (T_STEPS * 2));
        float4* rp = (float4*)(ret + (size_t)e * (T_STEPS * 2));
#pragma unroll
        for (int i = 0; i < 4; ++i) {
            float4 lv = lb[i];
            float4 ov;
            ov.x = outbuf[i * 4 + 0];
            ov.y = outbuf[i * 4 + 1];
            ov.z = outbuf[i * 4 + 2];
            ov.w = outbuf[i * 4 + 3];
            rp[i] = ov;
            float d0 = ov.x - lv.x, d1 = ov.y - lv.y;
            float d2 = ov.z - lv.z, d3 = ov.w - lv.w;
            lsum += d0 * d0 + d1 * d1 + d2 * d2 + d3 * d3;
        }
    }

    __shared__ float red[256];
    red[threadIdx.x] = lsum;
    __syncthreads();
#pragma unroll
    for (int s = 128; s > 0; s >>= 1) {
        if ((int)threadIdx.x < s) red[threadIdx.x] += red[threadIdx.x + s];
        __syncthreads();
    }
    if (threadIdx.x == 0) block_sum[blockIdx.x] = red[0];
}

__global__ __launch_bounds__(256) void loss_reduce_kernel(
    const float* __restrict__ block_sum, int nblocks,
    float* __restrict__ loss_out, float inv_n)
{
    __shared__ float red[256];
    float s = 0.0f;
    for (int i = (int)threadIdx.x; i < nblocks; i += 256) s += block_sum[i];
    red[threadIdx.x] = s;
    __syncthreads();
#pragma unroll
    for (int k = 128; k > 0; k >>= 1) {
        if ((int)threadIdx.x < k) red[threadIdx.x] += red[threadIdx.x + k];
        __syncthreads();
    }
    if (threadIdx.x == 0) *loss_out = red[0] * inv_n;
}

extern "C" void kernel_launch(void* const* d_in, const int* in_sizes, int n_in,
                              void* d_out, int out_size, void* d_ws, size_t ws_size,
                              hipStream_t stream) {
    const float* x_in   = (const float*)d_in[0];
    const float* labels = (const float*)d_in[1];
    const float* w_ih   = (const float*)d_in[2];
    const float* w_hh   = (const float*)d_in[3];
    const float* b_ih   = (const float*)d_in[4];

    const int B = in_sizes[0] / 2;                 // (B, H=2)
    const int N = B * T_STEPS * 2;                 // ret element count
    float* ret      = (float*)d_out;
    float* loss_out = ret + N;                     // last element of d_out
    float* bsum     = (float*)d_ws;

    // 256 threads = 8 waves; each wave handles 16 batch elems -> 128 per block.
    const int nblocks = B / 128;                   // B = 2^21 -> 16384, exact

    gru_fwd_kernel<<<nblocks, 256, 0, stream>>>(
        x_in, labels, w_ih, w_hh, b_ih, ret, bsum);
    loss_reduce_kernel<<<1, 256, 0, stream>>>(
        bsum, nblocks, loss_out, 1.0f / (float)N);
}